// MattingLaplacian_8589934792
// MI455X (gfx1250) — compile-verified
//
#include <hip/hip_runtime.h>
#include <stdint.h>

// ---------------------------------------------------------------------------
// Matting-Laplacian quadratic form, 512x512x3, 3x3 windows.
// Per window n, channel c:   contrib = |v|^2 - (1/9)*((sum v)^2 + y^T inv y)
// with y = Xc^T v,  inv = (cov + eps/9 I)^-1  (closed-form symmetric 3x3).
// - Tiles (18x18x3 f32) of target & style are DMA'd into LDS with the CDNA5
//   Tensor Data Mover (tensor_load_to_lds), synced via s_wait_tensorcnt.
// - Block/final reductions use V_WMMA_F32_16X16X4_F32 with an all-ones B
//   matrix (chained C accumulation) instead of an LDS barrier tree.
// ---------------------------------------------------------------------------

#define IMG_W 512
#define IMG_H 512
#define NWIN  (IMG_W - 2)     // 510 windows per dim
#define TB    16              // 16x16 windows per block
#define TILE  (TB + 2)        // 18x18 pixels per channel
#define EPS9  (1e-7f / 9.0f)

typedef unsigned int v4u __attribute__((ext_vector_type(4)));
typedef int          v8i __attribute__((ext_vector_type(8)));
typedef int          v4i __attribute__((ext_vector_type(4)));
typedef float        v2f __attribute__((ext_vector_type(2)));
typedef float        v8f __attribute__((ext_vector_type(8)));

// Issue one TDM load: 18x18x3 f32 tile starting at pixel (x0,y0) of a
// (3,512,512) CHW tensor -> LDS at byte offset lds_off (packed, x-major).
// tensor_dim0/1 are set to the remaining extent from the tile start so the
// TDM zero-fills out-of-bounds rows/cols at the image edge.
__device__ __forceinline__ void tdm_load_tile(const float* gbase,
                                              uint32_t lds_off,
                                              int x0, int y0) {
  uint64_t ga = (uint64_t)(uintptr_t)gbase +
                4ull * ((uint64_t)y0 * IMG_W + (uint64_t)x0);
  uint32_t ext_x = (uint32_t)(IMG_W - x0);   // valid elems in x from tile start
  uint32_t ext_y = (uint32_t)(IMG_H - y0);   // valid elems in y from tile start

  v4u g0;
  g0.x = 1u;                                   // count=1 (valid user D#)
  g0.y = lds_off;                              // LDS byte address
  g0.z = (uint32_t)(ga & 0xFFFFFFFFu);         // global_addr[31:0]
  g0.w = (uint32_t)((ga >> 32) & 0x01FFFFFFu)  // global_addr[56:32]
         | (2u << 30);                         // type=2 ("image")

  v8i g1;
  g1[0] = (int)(2u << 16);                     // data_size=2 -> 4 bytes
  g1[1] = (int)((ext_x & 0xFFFFu) << 16);      // tensor_dim0[15:0]
  g1[2] = (int)(((ext_x >> 16) & 0xFFFFu) |    // tensor_dim0[31:16]
                ((ext_y & 0xFFFFu) << 16));    // tensor_dim1[15:0]
  g1[3] = (int)(((ext_y >> 16) & 0xFFFFu) |    // tensor_dim1[31:16]
                ((uint32_t)TILE << 16));       // tile_dim0 = 18
  g1[4] = (int)((uint32_t)TILE |               // tile_dim1 = 18
                (3u << 16));                   // tile_dim2 = 3 channels
  g1[5] = (int)(uint32_t)IMG_W;                // tensor_dim0_stride[31:0] = 512
  g1[6] = 0;                                   // stride0[47:32] | stride1[15:0]
  g1[7] = (int)((uint32_t)(IMG_W * IMG_H) >> 16); // tensor_dim1_stride[47:16]

  v4i g2;
  g2[0] = 3;                                   // tensor_dim2 = 3 (channels)
  g2[1] = 0;                                   // tensor_dim3 (unused)
  g2[2] = 0;                                   // tensor_dim2_stride lo (unused)
  g2[3] = 0;                                   // stride hi | tile_dim3=0

  v4i g3 = {0, 0, 0, 0};                       // dims 3/4 unused
  v8i g4 = {0, 0, 0, 0, 0, 0, 0, 0};           // extra operand (6-arg form)

  __builtin_amdgcn_tensor_load_to_lds(g0, g1, g2, g3, g4, /*cpol=*/0);
}

// Wave-0-only: sum 256 f32 values in LDS using chained V_WMMA_F32_16X16X4_F32
// with an all-ones B matrix. Four 16x4 A-tiles (A layout: lane m holds K=0,1;
// lane m+16 holds K=2,3) accumulate through C; D[m][n] = sum of row m's 16
// values. Column 0 lives in lane 0 (M=0..7 in c[0..7]) and lane 16 (M=8..15).
// Result valid in lane 0. Must be called with all 32 lanes of the wave active.
__device__ __forceinline__ float wave0_wmma_sum256(const float* __restrict__ red,
                                                   int lane) {
  const int m  = lane & 15;
  const int kb = (lane < 16) ? 0 : 2;
  v2f b; b[0] = 1.0f; b[1] = 1.0f;
  v8f c = {};
#pragma unroll
  for (int t = 0; t < 4; ++t) {
    v2f a;
    a[0] = red[64 * t + 4 * m + kb];
    a[1] = red[64 * t + 4 * m + kb + 1];
    c = __builtin_amdgcn_wmma_f32_16x16x4_f32(
        /*neg_a=*/false, a, /*neg_b=*/false, b,
        /*c_mod=*/(short)0, c, /*reuse_a=*/false, /*reuse_b=*/false);
  }
  float s = ((c[0] + c[1]) + (c[2] + c[3])) + ((c[4] + c[5]) + (c[6] + c[7]));
  s += __shfl_down(s, 16, 32);   // lane0 += lane16 (rows 8..15)
  return s;
}

#define TIX(ch, y, x) ((ch) * (TILE * TILE) + (y) * TILE + (x))

__global__ void __launch_bounds__(TB * TB)
matting_laplacian_kernel(const float* __restrict__ target,
                         const float* __restrict__ style,
                         float* __restrict__ partials) {
  __shared__ float tileT[3 * TILE * TILE];  // 972 f32
  __shared__ float tileS[3 * TILE * TILE];  // 972 f32
  __shared__ float red[TB * TB];

  const int tx = threadIdx.x, ty = threadIdx.y;
  const int flat = ty * TB + tx;
  const int x0 = blockIdx.x * TB;
  const int y0 = blockIdx.y * TB;

  // Wave 0 issues both TDM tile loads; TDM ignores EXEC (one issue per wave).
  if (flat < 32) {
    tdm_load_tile(target, (uint32_t)(uintptr_t)tileT, x0, y0);
    tdm_load_tile(style,  (uint32_t)(uintptr_t)tileS, x0, y0);
    __builtin_amdgcn_s_wait_tensorcnt(0);
  }
  __syncthreads();

  const int wx = x0 + tx;
  const int wy = y0 + ty;
  float contrib = 0.0f;

  if (wx < NWIN && wy < NWIN) {
    float p[3][9], v[3][9];
#pragma unroll
    for (int ch = 0; ch < 3; ++ch)
#pragma unroll
      for (int dy = 0; dy < 3; ++dy)
#pragma unroll
        for (int dx = 0; dx < 3; ++dx) {
          const int j = dy * 3 + dx;
          p[ch][j] = tileT[TIX(ch, ty + dy, tx + dx)];
          v[ch][j] = tileS[TIX(ch, ty + dy, tx + dx)];
        }

    // Channel sums + 3x3 Gram matrix of the target window
    float S0 = 0.f, S1 = 0.f, S2 = 0.f;
    float G00 = 0.f, G01 = 0.f, G02 = 0.f, G11 = 0.f, G12 = 0.f, G22 = 0.f;
#pragma unroll
    for (int j = 0; j < 9; ++j) {
      const float a0 = p[0][j], a1 = p[1][j], a2 = p[2][j];
      S0 += a0; S1 += a1; S2 += a2;
      G00 = fmaf(a0, a0, G00); G01 = fmaf(a0, a1, G01); G02 = fmaf(a0, a2, G02);
      G11 = fmaf(a1, a1, G11); G12 = fmaf(a1, a2, G12); G22 = fmaf(a2, a2, G22);
    }
    const float inv9 = 1.0f / 9.0f;
    const float mu0 = S0 * inv9, mu1 = S1 * inv9, mu2 = S2 * inv9;

    // Regularized covariance (symmetric 3x3)
    const float a = G00 * inv9 - mu0 * mu0 + EPS9;
    const float b = G01 * inv9 - mu0 * mu1;
    const float c = G02 * inv9 - mu0 * mu2;
    const float d = G11 * inv9 - mu1 * mu1 + EPS9;
    const float e = G12 * inv9 - mu1 * mu2;
    const float f = G22 * inv9 - mu2 * mu2 + EPS9;

    // Closed-form symmetric inverse via adjugate
    const float c00 = d * f - e * e;
    const float c01 = c * e - b * f;
    const float c02 = b * e - c * d;
    const float det = a * c00 + b * c01 + c * c02;
    const float rdet = 1.0f / det;
    const float i00 = c00 * rdet, i01 = c01 * rdet, i02 = c02 * rdet;
    const float i11 = (a * f - c * c) * rdet;
    const float i12 = (b * c - a * e) * rdet;
    const float i22 = (a * d - b * b) * rdet;

#pragma unroll
    for (int ch = 0; ch < 3; ++ch) {
      float sv = 0.f, ssq = 0.f, y0v = 0.f, y1v = 0.f, y2v = 0.f;
#pragma unroll
      for (int j = 0; j < 9; ++j) {
        const float vj = v[ch][j];
        sv += vj;
        ssq = fmaf(vj, vj, ssq);
        y0v = fmaf(p[0][j], vj, y0v);
        y1v = fmaf(p[1][j], vj, y1v);
        y2v = fmaf(p[2][j], vj, y2v);
      }
      y0v -= mu0 * sv; y1v -= mu1 * sv; y2v -= mu2 * sv;
      const float q = y0v * y0v * i00 + y1v * y1v * i11 + y2v * y2v * i22 +
                      2.0f * (y0v * y1v * i01 + y0v * y2v * i02 + y1v * y2v * i12);
      contrib += ssq - inv9 * (sv * sv + q);
    }
  }

  // Block reduction: one barrier, then wave 0 sums 256 values on the matrix
  // pipe (4 chained WMMAs) instead of an 8-barrier LDS tree.
  red[flat] = contrib;
  __syncthreads();
  if (flat < 32) {
    const float tot = wave0_wmma_sum256(red, flat);
    if (flat == 0) partials[blockIdx.y * gridDim.x + blockIdx.x] = tot;
  }
}

__global__ void __launch_bounds__(256)
final_reduce_kernel(const float* __restrict__ partials, float* __restrict__ out,
                    int n) {
  __shared__ float sm[256];
  float s = 0.0f;
  for (int i = threadIdx.x; i < n; i += 256) s += partials[i];
  sm[threadIdx.x] = s;
  __syncthreads();
  if (threadIdx.x < 32) {
    const float tot = wave0_wmma_sum256(sm, threadIdx.x);
    if (threadIdx.x == 0) out[0] = tot;
  }
}

extern "C" void kernel_launch(void* const* d_in, const int* in_sizes, int n_in,
                              void* d_out, int out_size, void* d_ws, size_t ws_size,
                              hipStream_t stream) {
  (void)in_sizes; (void)n_in; (void)out_size; (void)ws_size;
  const float* target = (const float*)d_in[0];
  const float* style  = (const float*)d_in[1];
  float* partials = (float*)d_ws;      // 32*32 = 1024 f32 partials (4 KB)
  float* out = (float*)d_out;

  const int nb = (NWIN + TB - 1) / TB; // 32
  dim3 grid(nb, nb), block(TB, TB);
  matting_laplacian_kernel<<<grid, block, 0, stream>>>(target, style, partials);
  final_reduce_kernel<<<1, 256, 0, stream>>>(partials, out, nb * nb);
}